// Pool_26517128085982
// MI455X (gfx1250) — compile-verified
//
#include <hip/hip_runtime.h>
#include <cstdint>

// 2x2 mean pool, stride 2, over (16,64,512,512) f32 -> (16,64,256,256) f32.
// Memory-bound: ~1.34 GB traffic -> ~58us floor @ 23.3 TB/s.
// Strategy: b128-only transactions, non-temporal hints (stream-once data >> 192MB L2),
// and CDNA5 async global->LDS staging (ASYNCcnt) for deep MLP per wave.

typedef float v4f __attribute__((ext_vector_type(4)));
typedef int   v4i __attribute__((ext_vector_type(4)));

#define AS1 __attribute__((address_space(1)))
#define AS3 __attribute__((address_space(3)))

#if __has_builtin(__builtin_amdgcn_global_load_async_to_lds_b128)
#define HAVE_ASYNC_LDS 1
#else
#define HAVE_ASYNC_LDS 0
#endif

// gfx12-family CPol immediate: TH in bits [2:0], scope in bits [4:3].
// TH_LOAD_NT = 1, scope WGP = 0 -> stream-once load, minimal cache pollution.
#define CPOL_LOAD_NT 1

static constexpr int W_IN = 512;
static constexpr int H_IN = 512;
static constexpr int ROWS_PER_BLK = 8;                    // input rows per block
static constexpr int TILE_FLOATS = ROWS_PER_BLK * W_IN;   // 4096 floats = 16 KB
static constexpr int THREADS = 256;                       // 8 waves (wave32)

__global__ __launch_bounds__(THREADS) void pool2x2_kernel(
    const float* __restrict__ x, const float* __restrict__ kern,
    float* __restrict__ out)
{
    const int t   = threadIdx.x;
    const int bid = blockIdx.x;
    const int img = bid >> 6;   // 1024 images (B*C)
    const int rg  = bid & 63;   // 64 row-groups of 8 input rows

    // general 2x2 kernel weights (reference uses 0.25 each)
    const float k00 = kern[0], k01 = kern[1], k10 = kern[2], k11 = kern[3];

    // 8 full rows of one image are contiguous: one 16 KB linear tile
    const float* tile = x + (size_t)img * (size_t)(H_IN * W_IN)
                          + (size_t)rg * (size_t)TILE_FLOATS;

    const int r = t >> 6;   // output row within block, 0..3
    const int c = t & 63;   // output col group, 4 outputs each

    v4f a0, a1, b0, b1;

#if HAVE_ASYNC_LDS
    __shared__ float lds[TILE_FLOATS];
    // 1024 x 16B chunks, 4 per thread: fire-and-forget async DMA into LDS.
    #pragma unroll
    for (int i = 0; i < 4; ++i) {
        const int ci = t + i * THREADS;
        __builtin_amdgcn_global_load_async_to_lds_b128(
            (AS1 v4i*)(tile + ci * 4),
            (AS3 v4i*)(lds + ci * 4),
            /*imm offset*/0, /*cpol*/CPOL_LOAD_NT);
    }
#if __has_builtin(__builtin_amdgcn_s_wait_asynccnt)
    __builtin_amdgcn_s_wait_asynccnt(0);
#else
    asm volatile("s_wait_asynccnt 0" ::: "memory");
#endif
    __syncthreads();

    const float* r0 = lds + (2 * r) * W_IN + c * 8;
    const float* r1 = r0 + W_IN;
    a0 = *(const v4f*)(r0);
    a1 = *(const v4f*)(r0 + 4);
    b0 = *(const v4f*)(r1);
    b1 = *(const v4f*)(r1 + 4);
#else
    // Fallback: direct non-temporal b128 loads (no LDS round trip).
    const float* r0 = tile + (2 * r) * W_IN + c * 8;
    const float* r1 = r0 + W_IN;
    a0 = __builtin_nontemporal_load((const v4f*)(r0));
    a1 = __builtin_nontemporal_load((const v4f*)(r0 + 4));
    b0 = __builtin_nontemporal_load((const v4f*)(r1));
    b1 = __builtin_nontemporal_load((const v4f*)(r1 + 4));
#endif

    v4f o;
    o.x = k00 * a0.x + k01 * a0.y + k10 * b0.x + k11 * b0.y;
    o.y = k00 * a0.z + k01 * a0.w + k10 * b0.z + k11 * b0.w;
    o.z = k00 * a1.x + k01 * a1.y + k10 * b1.x + k11 * b1.y;
    o.w = k00 * a1.z + k01 * a1.w + k10 * b1.z + k11 * b1.w;

    float* op = out + (size_t)img * (size_t)(256 * 256)
                    + (size_t)(rg * 4 + r) * 256 + (size_t)c * 4;
    __builtin_nontemporal_store(o, (v4f*)op);
}

extern "C" void kernel_launch(void* const* d_in, const int* in_sizes, int n_in,
                              void* d_out, int out_size, void* d_ws, size_t ws_size,
                              hipStream_t stream) {
    const float* x = (const float*)d_in[0];   // (16,64,512,512) f32
    const float* k = (const float*)d_in[1];   // (2,2) f32 row-major
    float* out = (float*)d_out;               // (16,64,256,256) f32
    (void)in_sizes; (void)n_in; (void)out_size; (void)d_ws; (void)ws_size;

    dim3 grid(65536);   // 1024 images * 64 row-groups
    dim3 block(THREADS);
    hipLaunchKernelGGL(pool2x2_kernel, grid, block, 0, stream, x, k, out);
}